// GraphConv_77919296684699
// MI455X (gfx1250) — compile-verified
//
#include <hip/hip_runtime.h>
#include <math.h>

#define N_NODE 100000
#define N_EDGE 1600000
#define D_IN   128
#define D_OUT  128
#define D_EDGE 32
#define EPSF   1e-10f

typedef __attribute__((ext_vector_type(2))) float v2f;
typedef __attribute__((ext_vector_type(8))) float v8f;

// ---------------------------------------------------------------------------
// K0: deg_in/deg_out <- 1.0 (self loop), accumulator (= d_out) <- 0
// ---------------------------------------------------------------------------
__global__ void k_init(float* __restrict__ acc,
                       float* __restrict__ deg_in,
                       float* __restrict__ deg_out) {
    int i = blockIdx.x * blockDim.x + threadIdx.x;
    if (i < N_NODE * D_IN) acc[i] = 0.0f;
    if (i < N_NODE) { deg_in[i] = 1.0f; deg_out[i] = 1.0f; }
}

// ---------------------------------------------------------------------------
// K1: weighted degree scatter (scalar f32 atomics, L2-resident arrays)
// ---------------------------------------------------------------------------
__global__ void k_deg(const int* __restrict__ el,
                      const float* __restrict__ ew,
                      float* __restrict__ deg_in,
                      float* __restrict__ deg_out) {
    int e = blockIdx.x * blockDim.x + threadIdx.x;
    if (e >= N_EDGE) return;
    int s = el[2 * e + 0];
    int d = el[2 * e + 1];
    float w = ew[e];
    atomicAdd(&deg_in[s], w);
    atomicAdd(&deg_out[d], w);
}

// ---------------------------------------------------------------------------
// K2: fused edge GEMM (WMMA f32 16x16x4) + normalized scatter.
//   Block = 256 threads (8 wave32), 16 edges per block.
//   E_tile(16x128) = F_tile(16x32) @ W_edge^T(32x128) + b_edge, in LDS only.
//   Then update[dst] += w * (x[src] + E_row), 16 threads per edge, coalesced.
// ---------------------------------------------------------------------------
__global__ __launch_bounds__(256)
void k_edge(const float* __restrict__ x,
            const int*   __restrict__ el,
            const float* __restrict__ ew,
            const float* __restrict__ ef,
            const float* __restrict__ W_edge,   // (D_IN=128, D_EDGE=32) row-major
            const float* __restrict__ b_edge,   // (D_IN)
            const float* __restrict__ deg_in,
            const float* __restrict__ deg_out,
            float* __restrict__ acc) {
    __shared__ float sF[16 * D_EDGE];   // 2 KB: 16 edges x 32 features
    __shared__ float sE[16 * D_IN];     // 8 KB: 16 edges x 128 transformed

    const int tid = threadIdx.x;
    const int e0  = blockIdx.x * 16;    // N_EDGE/16 = 100000 blocks, exact

    // Stage feature tile (coalesced, 2 floats/thread)
    for (int i = tid; i < 16 * D_EDGE; i += 256) {
        int m = i >> 5, k = i & 31;
        sF[i] = ef[(size_t)(e0 + m) * D_EDGE + k];
    }
    __syncthreads();

    const int  wave = tid >> 5;
    const int  lane = tid & 31;
    const int  lh   = lane & 15;        // M / N index within half-wave
    const bool hiK  = (lane >= 16);     // high lanes carry K+2/K+3 (A/B) and M+8 (C/D)
    const int  n0   = wave * 16;        // this wave's 16-column output slice

    // D = A*B + C over K=32 in 8 steps of K=4 (v_wmma_f32_16x16x4_f32)
    v8f c = {};
#pragma unroll
    for (int kk = 0; kk < D_EDGE / 4; ++kk) {
        const int kb = kk * 4 + (hiK ? 2 : 0);
        v2f a, b;
        // A (16x4 f32): lane L<16 -> {A[M=L][K0], A[M=L][K1]}, L>=16 -> K2/K3
        a.x = sF[lh * D_EDGE + kb + 0];
        a.y = sF[lh * D_EDGE + kb + 1];
        // B (4x16 f32): B[k][n] = W_edge[n][k], same half-lane K split
        b.x = W_edge[(n0 + lh) * D_EDGE + kb + 0];
        b.y = W_edge[(n0 + lh) * D_EDGE + kb + 1];
        c = __builtin_amdgcn_wmma_f32_16x16x4_f32(
                /*neg_a=*/false, a, /*neg_b=*/false, b,
                /*c_mod=*/(short)0, c, /*reuse_a=*/false, /*reuse_b=*/false);
    }

    // C/D layout: VGPR r holds M=r (lanes 0-15) and M=r+8 (lanes 16-31)
    const float bias = b_edge[n0 + lh];
    const int   mr   = hiK ? 8 : 0;
#pragma unroll
    for (int r = 0; r < 8; ++r)
        sE[(mr + r) * D_IN + (n0 + lh)] = c[r] + bias;
    __syncthreads();

    // Scatter: 16 threads per edge, each covers 8 columns (stride-16 -> each
    // group of 16 consecutive threads hits 16 consecutive floats = 64B).
    const int   m  = tid >> 4;          // edge in tile
    const int   cb = tid & 15;
    const int   e  = e0 + m;
    const int   s  = el[2 * e + 0];
    const int   d  = el[2 * e + 1];
    const float w  = ew[e] / (sqrtf(deg_in[s] * deg_out[d]) + EPSF);
    const float* xs = x   + (size_t)s * D_IN;
    float*       ad = acc + (size_t)d * D_IN;
#pragma unroll
    for (int j = 0; j < 8; ++j) {
        int col = cb + j * 16;
        atomicAdd(&ad[col], w * (xs[col] + sE[m * D_IN + col]));
    }
}

// ---------------------------------------------------------------------------
// K3: self-loop add + final GEMM + bias + ReLU, in place on d_out.
//   Block = 256 threads, 16 nodes per block (100000/16 = 6250 blocks, exact).
//   h = update + (1/(sqrt(deg_in*deg_out)+eps)) * x   (staged in LDS)
//   out = relu(h @ W_lin^T + b_lin)
// ---------------------------------------------------------------------------
__global__ __launch_bounds__(256)
void k_node(const float* __restrict__ x,
            const float* __restrict__ W_lin,   // (D_OUT=128, D_IN=128) row-major
            const float* __restrict__ b_lin,
            const float* __restrict__ deg_in,
            const float* __restrict__ deg_out,
            float* __restrict__ out) {         // accumulator in, result out
    __shared__ float sH[16 * D_IN];            // 8 KB
    __shared__ float sW[16];

    const int tid = threadIdx.x;
    const int i0  = blockIdx.x * 16;

    if (tid < 16) {
        int n = i0 + tid;
        sW[tid] = 1.0f / (sqrtf(deg_in[n] * deg_out[n]) + EPSF);
    }
    __syncthreads();

    // All reads of this block's 16 rows complete before any write-back.
    for (int i = tid; i < 16 * D_IN; i += 256) {
        int m = i >> 7, k = i & 127;
        size_t g = (size_t)(i0 + m) * D_IN + k;
        sH[i] = out[g] + sW[m] * x[g];
    }
    __syncthreads();

    const int  wave = tid >> 5;
    const int  lane = tid & 31;
    const int  lh   = lane & 15;
    const bool hiK  = (lane >= 16);
    const int  n0   = wave * 16;

    v8f c = {};
#pragma unroll
    for (int kk = 0; kk < D_IN / 4; ++kk) {
        const int kb = kk * 4 + (hiK ? 2 : 0);
        v2f a, b;
        a.x = sH[lh * D_IN + kb + 0];
        a.y = sH[lh * D_IN + kb + 1];
        b.x = W_lin[(n0 + lh) * D_IN + kb + 0];
        b.y = W_lin[(n0 + lh) * D_IN + kb + 1];
        c = __builtin_amdgcn_wmma_f32_16x16x4_f32(
                false, a, false, b, (short)0, c, false, false);
    }

    const float bias = b_lin[n0 + lh];
    const int   mr   = hiK ? 8 : 0;
#pragma unroll
    for (int r = 0; r < 8; ++r) {
        float v = c[r] + bias;
        out[(size_t)(i0 + mr + r) * D_IN + (n0 + lh)] = v > 0.0f ? v : 0.0f;
    }
}

// ---------------------------------------------------------------------------
extern "C" void kernel_launch(void* const* d_in, const int* in_sizes, int n_in,
                              void* d_out, int out_size, void* d_ws, size_t ws_size,
                              hipStream_t stream) {
    const float* x      = (const float*)d_in[0];
    const int*   el     = (const int*)  d_in[1];
    const float* ew     = (const float*)d_in[2];
    const float* ef     = (const float*)d_in[3];
    const float* W_lin  = (const float*)d_in[4];
    const float* b_lin  = (const float*)d_in[5];
    const float* W_edge = (const float*)d_in[6];
    const float* b_edge = (const float*)d_in[7];

    float* out      = (float*)d_out;         // doubles as the update accumulator
    float* deg_in   = (float*)d_ws;          // N_NODE floats
    float* deg_outp = deg_in + N_NODE;       // N_NODE floats (800 KB total ws)

    k_init<<<(N_NODE * D_IN + 255) / 256, 256, 0, stream>>>(out, deg_in, deg_outp);
    k_deg <<<(N_EDGE + 255) / 256,        256, 0, stream>>>(el, ew, deg_in, deg_outp);
    k_edge<<<N_EDGE / 16,                 256, 0, stream>>>(x, el, ew, ef, W_edge, b_edge,
                                                           deg_in, deg_outp, out);
    k_node<<<N_NODE / 16,                 256, 0, stream>>>(x, W_lin, b_lin,
                                                           deg_in, deg_outp, out);
}